// MetaLearnerRegressionCol_20229295964905
// MI455X (gfx1250) — compile-verified
//
#include <hip/hip_runtime.h>
#include <hip/hip_bf16.h>

// ---------------------------------------------------------------------------
// Problem constants (from reference): T=100, N=100, D=1024, H=8192, lr=0.01
// ---------------------------------------------------------------------------
#define TT 100
#define NN 100
#define DD 1024
#define HH 8192
#define LR 0.01f

typedef __attribute__((ext_vector_type(2))) float v2f;
typedef __attribute__((ext_vector_type(8))) float v8f;

// ---------------------------------------------------------------------------
// Kernel 1: Hcat[m, j] = tanh( dot(x_m, W[:,j]) + b[j] ),  m in [0,200)
//   rows 0..99 from x_traj, rows 100..199 from x_rand.
//   One wave computes one 16x16 tile via V_WMMA_F32_16X16X4_F32 (K-step 4).
//   Out-of-range M rows (200..207) are fed row-199 data (finite garbage);
//   D[m,*] depends only on A[m,*], and those rows are never stored, so no
//   masking is needed in the hot loop.
// ---------------------------------------------------------------------------
__global__ void __launch_bounds__(256)
k_hidden(const float* __restrict__ xt, const float* __restrict__ xr,
         const float* __restrict__ W,  const float* __restrict__ bias,
         float* __restrict__ Hcat) {
  const int lane  = threadIdx.x & 31;
  const int wave  = threadIdx.x >> 5;
  const int col   = blockIdx.x * 128 + wave * 16 + (lane & 15); // N index
  const int mbase = blockIdx.y * 16;
  const int row   = mbase + (lane & 15);                        // A row (M)
  const int krel  = (lane >> 4) << 1;                           // 0 or 2

  const int srow = (row < TT + NN) ? row : (TT + NN - 1);       // clamp only
  const float* arow = (srow < TT) ? (xt + (size_t)srow * DD)
                                  : (xr + (size_t)(srow - TT) * DD);
  v8f acc = {};
  for (int k = 0; k < DD; k += 4) {
    v2f a, b;
    a.x = arow[k + krel];
    a.y = arow[k + krel + 1];
    b.x = W[(k + krel)     * HH + col];
    b.y = W[(k + krel + 1) * HH + col];
    if (k + 8 < DD)  // stream W through L2 (global_prefetch_b8)
      __builtin_prefetch(&W[(k + 8) * HH + col], 0, 1);
    acc = __builtin_amdgcn_wmma_f32_16x16x4_f32(
        false, a, false, b, (short)0, acc, false, false);
  }
  const float bb  = bias[col];
  const int   mhi = (lane >> 4) << 3;  // lanes 16..31 hold M+8 rows
#pragma unroll
  for (int r = 0; r < 8; ++r) {
    const int mr = mbase + mhi + r;
    if (mr < TT + NN) Hcat[(size_t)mr * HH + col] = tanhf(acc[r] + bb);
  }
}

// ---------------------------------------------------------------------------
// Kernel 2: sequential trajectory scan (single 1024-thread workgroup).
//   Forward:  per step t, y = dot(w, h_t) (wave32 shfl_xor butterfly + one
//             32-entry LDS stage -> 3 barriers/step), err, w update, store
//             alpha[t,j] = 1 - lr*h^2, beta[t,j] = lr*dact*(err - h*w_new).
//   Backward: gamma[t,j] = beta[t,j] * prod_{s>t} alpha[s,j] (in place over
//             beta), TW_b[j] = sum_t gamma[t,j].  Also writes w_f.
//   Thread tid owns columns j = tid + r*1024, r=0..7 (coalesced).
// ---------------------------------------------------------------------------
__global__ void __launch_bounds__(1024)
k_traj(const float* __restrict__ Hcat, const float* __restrict__ ytraj,
       const float* __restrict__ w0, float* __restrict__ alphaB,
       float* __restrict__ betaB, float* __restrict__ wf,
       float* __restrict__ twb) {
  __shared__ float red[32];
  const int tid  = threadIdx.x;
  const int lane = tid & 31;
  const int wave = tid >> 5;
  float w[8];
#pragma unroll
  for (int r = 0; r < 8; ++r) w[r] = w0[tid + r * 1024];

  for (int t = 0; t < TT; ++t) {
    float h[8];
    float local = 0.0f;
#pragma unroll
    for (int r = 0; r < 8; ++r) {
      h[r] = Hcat[(size_t)t * HH + tid + r * 1024];
      local += w[r] * h[r];
    }
    // wave32 butterfly reduce (no barrier), then 32-wave LDS stage
#pragma unroll
    for (int off = 16; off > 0; off >>= 1)
      local += __shfl_xor(local, off, 32);
    if (lane == 0) red[wave] = local;
    __syncthreads();
    if (wave == 0) {
      float u = red[lane];
#pragma unroll
      for (int off = 16; off > 0; off >>= 1)
        u += __shfl_xor(u, off, 32);
      if (lane == 0) red[0] = u;
    }
    __syncthreads();
    const float y = red[0];
    __syncthreads();  // protect red[0] before next step's writes
    const float err = ytraj[t] - y;
#pragma unroll
    for (int r = 0; r < 8; ++r) {
      const float hh    = h[r];
      const float dact  = 1.0f - hh * hh;
      const float wn    = w[r] + LR * err * hh;
      const float alpha = 1.0f - LR * hh * hh;
      const float beta  = LR * dact * (err - hh * wn);
      alphaB[(size_t)t * HH + tid + r * 1024] = alpha;
      betaB [(size_t)t * HH + tid + r * 1024] = beta;
      w[r] = wn;
    }
  }
  float P[8], tb[8];
#pragma unroll
  for (int r = 0; r < 8; ++r) {
    P[r] = 1.0f;
    tb[r] = 0.0f;
    wf[tid + r * 1024] = w[r];
  }
  for (int t = TT - 1; t >= 0; --t) {
#pragma unroll
    for (int r = 0; r < 8; ++r) {
      const size_t idx = (size_t)t * HH + tid + r * 1024;
      const float g = betaB[idx] * P[r];
      betaB[idx] = g;  // gamma in place
      tb[r] += g;
      P[r] *= alphaB[idx];
    }
  }
#pragma unroll
  for (int r = 0; r < 8; ++r) twb[tid + r * 1024] = tb[r];
}

// ---------------------------------------------------------------------------
// Kernel 3: err[n] = y_rand[n] - dot(w_f, Hrand[n,:])   (one block per n)
// ---------------------------------------------------------------------------
__global__ void __launch_bounds__(256)
k_err(const float* __restrict__ Hcat, const float* __restrict__ wf,
      const float* __restrict__ yrand, float* __restrict__ errv) {
  __shared__ float red[8];
  const int n = blockIdx.x, tid = threadIdx.x;
  const int lane = tid & 31, wave = tid >> 5;
  const float* hrow = Hcat + (size_t)(TT + n) * HH;
  float local = 0.0f;
  for (int j = tid; j < HH; j += 256) local += wf[j] * hrow[j];
#pragma unroll
  for (int off = 16; off > 0; off >>= 1) local += __shfl_xor(local, off, 32);
  if (lane == 0) red[wave] = local;
  __syncthreads();
  if (tid == 0) {
    float u = 0.0f;
#pragma unroll
    for (int i = 0; i < 8; ++i) u += red[i];
    errv[n] = yrand[n] - u;
  }
}

// ---------------------------------------------------------------------------
// Kernel 4: out[0] = (sum_n err^2) / N   (deterministic single-block reduce)
// ---------------------------------------------------------------------------
__global__ void __launch_bounds__(128)
k_loss(const float* __restrict__ errv, float* __restrict__ out) {
  __shared__ float red[128];
  const int tid = threadIdx.x;
  const float v = (tid < NN) ? errv[tid] : 0.0f;
  red[tid] = v * v;
  __syncthreads();
  for (int s = 64; s > 0; s >>= 1) {
    if (tid < s) red[tid] += red[tid + s];
    __syncthreads();
  }
  if (tid == 0) out[0] = red[0] / (float)NN;
}

// ---------------------------------------------------------------------------
// Kernel 5: per column j:  A[n,j] = -err_n * w_f[j] * (1-h^2),
//   s_j = sum_n(-err_n h),  gb[j] = sum_n A[n,j] + TW_b[j]*s_j,
//   gamma[t,j] *= s_j  (turns gamma buffer into Gs for the final GEMM).
// ---------------------------------------------------------------------------
__global__ void __launch_bounds__(256)
k_fields(const float* __restrict__ Hcat, const float* __restrict__ errv,
         const float* __restrict__ wf, const float* __restrict__ twb,
         float* __restrict__ Amat, float* __restrict__ gammaB,
         float* __restrict__ gb) {
  const int j = blockIdx.x * blockDim.x + threadIdx.x;
  const float wfj = wf[j];
  float s = 0.0f, gsum = 0.0f;
  for (int n = 0; n < NN; ++n) {
    const float h = Hcat[(size_t)(TT + n) * HH + j];
    const float e = errv[n];
    const float a = -e * wfj * (1.0f - h * h);
    Amat[(size_t)n * HH + j] = a;
    gsum += a;
    s += -e * h;
  }
  gb[j] = gsum + twb[j] * s;
  for (int t = 0; t < TT; ++t) gammaB[(size_t)t * HH + j] *= s;
}

// ---------------------------------------------------------------------------
// Kernel 6: gW = X_rand^T @ A + X_traj^T @ Gs   ([1024,8192], two K=100 passes)
//   Same WMMA tiling as k_hidden; A-operand reads are transposed accesses of
//   the row-major [100,1024] inputs (coalesced across lanes within a k-row).
// ---------------------------------------------------------------------------
__global__ void __launch_bounds__(256)
k_grad(const float* __restrict__ Xr, const float* __restrict__ Amat,
       const float* __restrict__ Xt, const float* __restrict__ Gs,
       float* __restrict__ gW) {
  const int lane  = threadIdx.x & 31;
  const int wave  = threadIdx.x >> 5;
  const int col   = blockIdx.x * 128 + wave * 16 + (lane & 15);
  const int mbase = blockIdx.y * 16;
  const int drow  = mbase + (lane & 15);
  const int krel  = (lane >> 4) << 1;
  v8f acc = {};
  for (int k = 0; k < NN; k += 4) {  // pass 1: X_rand^T @ A
    v2f a, b;
    a.x = Xr[(k + krel)     * DD + drow];
    a.y = Xr[(k + krel + 1) * DD + drow];
    b.x = Amat[(size_t)(k + krel)     * HH + col];
    b.y = Amat[(size_t)(k + krel + 1) * HH + col];
    acc = __builtin_amdgcn_wmma_f32_16x16x4_f32(
        false, a, false, b, (short)0, acc, false, false);
  }
  for (int k = 0; k < TT; k += 4) {  // pass 2: X_traj^T @ (gamma*s)
    v2f a, b;
    a.x = Xt[(k + krel)     * DD + drow];
    a.y = Xt[(k + krel + 1) * DD + drow];
    b.x = Gs[(size_t)(k + krel)     * HH + col];
    b.y = Gs[(size_t)(k + krel + 1) * HH + col];
    acc = __builtin_amdgcn_wmma_f32_16x16x4_f32(
        false, a, false, b, (short)0, acc, false, false);
  }
  const int mhi = (lane >> 4) << 3;
#pragma unroll
  for (int r = 0; r < 8; ++r)
    gW[(size_t)(mbase + mhi + r) * HH + col] = acc[r];
}

// ---------------------------------------------------------------------------
// Host launcher
// ---------------------------------------------------------------------------
extern "C" void kernel_launch(void* const* d_in, const int* in_sizes, int n_in,
                              void* d_out, int out_size, void* d_ws,
                              size_t ws_size, hipStream_t stream) {
  (void)in_sizes; (void)n_in; (void)out_size; (void)ws_size;
  const float* xt = (const float*)d_in[0];  // [100,1024]
  const float* yt = (const float*)d_in[1];  // [100,1]
  const float* xr = (const float*)d_in[2];  // [100,1024]
  const float* yr = (const float*)d_in[3];  // [100,1]
  const float* W  = (const float*)d_in[4];  // [1024,8192]
  const float* b  = (const float*)d_in[5];  // [8192]
  const float* w0 = (const float*)d_in[6];  // [8192]

  float* out = (float*)d_out;          // [0]=loss, [1..1+D*H)=gW, tail=gb
  float* gW  = out + 1;
  float* gb  = out + 1 + (size_t)DD * HH;

  float* ws     = (float*)d_ws;        // ~16.4 MB of f32 scratch (L2 resident)
  float* Hcat   = ws;                             // 200*8192
  float* alphaB = Hcat + (size_t)(TT + NN) * HH;  // 100*8192
  float* betaB  = alphaB + (size_t)TT * HH;       // 100*8192 (-> gamma -> Gs)
  float* Amat   = betaB + (size_t)TT * HH;        // 100*8192
  float* wf     = Amat + (size_t)NN * HH;         // 8192
  float* twb    = wf + HH;                        // 8192
  float* errv   = twb + HH;                       // 128

  k_hidden<<<dim3(HH / 128, (TT + NN + 15) / 16), 256, 0, stream>>>(
      xt, xr, W, b, Hcat);
  k_traj<<<1, 1024, 0, stream>>>(Hcat, yt, w0, alphaB, betaB, wf, twb);
  k_err<<<NN, 256, 0, stream>>>(Hcat, wf, yr, errv);
  k_loss<<<1, 128, 0, stream>>>(errv, out);
  k_fields<<<HH / 256, 256, 0, stream>>>(Hcat, errv, wf, twb, Amat, betaB, gb);
  k_grad<<<dim3(HH / 128, DD / 16), 256, 0, stream>>>(xr, Amat, xt, betaB, gW);
}